// LipschitzMargin_63410897158181
// MI455X (gfx1250) — compile-verified
//
#include <hip/hip_runtime.h>
#include <math.h>

typedef __attribute__((ext_vector_type(2))) float v2f;
typedef __attribute__((ext_vector_type(8))) float v8f;

#define NB 512      // batch rows (GEMM M)
#define NC 1000     // classes   (GEMM N)
#define NCP 1024    // padded classes
#define ND 512      // features  (GEMM K)
#define EPS_CERT 0.3f
#define EPS_SQRT 1e-10f

// ---- workspace layout (float offsets) ----
// kWp    : [ND][NCP]  at 0               (2 MB)  zero-padded kW
// At     : [ND][NB]   at 524288          (1 MB)  gathered A, K-major
// colnorm: [NCP]      at 786432
// ymax   : [NB]       at 787456
// jidx   : [NB] (int) at 787968
// partial: [NB][16]   at 788480
#define WS_KWP     0
#define WS_AT      524288
#define WS_COLNORM 786432
#define WS_YMAX    787456
#define WS_JIDX    787968
#define WS_PARTIAL 788480

// Kernel 1: pad kW -> kWp, compute column norms.
__global__ void lm_prep(const float* __restrict__ kW, float* __restrict__ kWp,
                        float* __restrict__ colnorm) {
    int i = blockIdx.x * blockDim.x + threadIdx.x;   // 0..NCP-1
    if (i >= NCP) return;
    float acc = 0.f;
    for (int d = 0; d < ND; ++d) {
        float v = (i < NC) ? kW[d * NC + i] : 0.f;
        kWp[d * NCP + i] = v;
        acc = fmaf(v, v, acc);
    }
    colnorm[i] = acc;
}

// Kernel 2: per-row argmax (first-occurrence tie-break), copy y into out.
__global__ void lm_argmax(const float* __restrict__ y, float* __restrict__ out,
                          float* __restrict__ ymax, int* __restrict__ jidx) {
    __shared__ float sval[256];
    __shared__ int   sidx[256];
    int b = blockIdx.x;
    int tid = threadIdx.x;
    const float* yr = y + (size_t)b * NC;

    float best = -__builtin_inff();
    int bi = 0;
    for (int i = tid; i < NC; i += 256) {
        float v = yr[i];
        if (v > best) { best = v; bi = i; }
    }
    sval[tid] = best; sidx[tid] = bi;
    __syncthreads();
    for (int s = 128; s > 0; s >>= 1) {
        if (tid < s) {
            float vo = sval[tid + s]; int io = sidx[tid + s];
            if (vo > sval[tid] || (vo == sval[tid] && io < sidx[tid])) {
                sval[tid] = vo; sidx[tid] = io;
            }
        }
        __syncthreads();
    }
    if (tid == 0) { ymax[b] = sval[0]; jidx[b] = sidx[0]; }
    // copy logits into output row (C+1 wide)
    float* orow = out + (size_t)b * (NC + 1);
    for (int i = tid; i < NC; i += 256) orow[i] = yr[i];
}

// Kernel 3: gather A in K-major layout: At[k*NB + m] = kWp[k*NCP + jidx[m]]
__global__ void lm_gatherA(const float* __restrict__ kWp,
                           const int* __restrict__ jidx,
                           float* __restrict__ At) {
    int idx = blockIdx.x * blockDim.x + threadIdx.x;   // 0..ND*NB-1
    int m = idx & (NB - 1);
    int k = idx >> 9;                                   // /NB
    At[idx] = kWp[k * NCP + jidx[m]];
}

// Kernel 4: GEMM (WMMA f32 16x16x4) + fused certification epilogue.
// grid (8,8): x = N-tile (128 cols), y = M-tile (64 rows); 256 threads = 8 waves.
__global__ void lm_gemm(const float* __restrict__ At,    // [ND][NB]
                        const float* __restrict__ kWp,   // [ND][NCP]
                        const float* __restrict__ colnorm,
                        const float* __restrict__ y,
                        const float* __restrict__ ymax,
                        const int*   __restrict__ jidx,
                        float* __restrict__ partial) {   // [NB][16]
    const int tid  = threadIdx.x;
    const int lane = tid & 31;
    const int w    = tid >> 5;      // 0..7
    const int mw   = w >> 1;        // 0..3
    const int nw   = w & 1;         // 0..1
    const int half = lane >> 4;     // 0|1
    const int lm   = lane & 15;

    const int mrow0 = blockIdx.y * 64 + mw * 16;
    const int ncol0 = blockIdx.x * 128 + nw * 64;

    v8f acc[4] = {v8f{}, v8f{}, v8f{}, v8f{}};

    const float* Ab = At  + mrow0 + lm;       // + kk*NB
    const float* Bb = kWp + ncol0 + lm;       // + kk*NCP + nb*16

    for (int k = 0; k < ND; k += 4) {
        const int kk = k + 2 * half;
        // A fragment: lane lm holds rows M=mrow0+lm, K = kk, kk+1
        v2f a;
        a.x = Ab[kk * NB];
        a.y = Ab[(kk + 1) * NB];
        const float* bp0 = Bb + kk * NCP;
        const float* bp1 = Bb + (kk + 1) * NCP;
#pragma unroll
        for (int nb = 0; nb < 4; ++nb) {
            v2f bf;
            bf.x = bp0[nb * 16];
            bf.y = bp1[nb * 16];
            acc[nb] = __builtin_amdgcn_wmma_f32_16x16x4_f32(
                false, a, false, bf, (short)0, acc[nb], false, false);
        }
    }

    // Epilogue: cand = (y==ymax) ? -inf : y + 0.3*sqrt(cn_j + cn_i - 2*dot + eps)
    // C/D layout: VGPR r -> M = mrow0 + r + 8*half, N = ncol0 + nb*16 + lm.
    const float ninf = -__builtin_inff();
#pragma unroll
    for (int r = 0; r < 8; ++r) {
        const int M   = mrow0 + r + 8 * half;
        const float cnj = colnorm[jidx[M]];
        const float ym  = ymax[M];
        float v = ninf;
#pragma unroll
        for (int nb = 0; nb < 4; ++nb) {
            const int N = ncol0 + nb * 16 + lm;
            if (N < NC) {
                const float dot = acc[nb][r];
                float k2 = fmaxf(cnj + colnorm[N] - 2.f * dot, 0.f);
                float Kd = sqrtf(k2 + EPS_SQRT);
                float yv = y[(size_t)M * NC + N];
                float cand = (yv == ym) ? ninf : fmaf(EPS_CERT, Kd, yv);
                v = fmaxf(v, cand);
            }
        }
        // butterfly max within each 16-lane half (rows are per-half)
        v = fmaxf(v, __shfl_xor(v, 1, 16));
        v = fmaxf(v, __shfl_xor(v, 2, 16));
        v = fmaxf(v, __shfl_xor(v, 4, 16));
        v = fmaxf(v, __shfl_xor(v, 8, 16));
        if (lm == 0) partial[M * 16 + blockIdx.x * 2 + nw] = v;
    }
}

// Kernel 5: reduce 16 partials per row -> out[b][NC]
__global__ void lm_ybot(const float* __restrict__ partial, float* __restrict__ out) {
    int b = blockIdx.x * blockDim.x + threadIdx.x;
    if (b >= NB) return;
    float m = -__builtin_inff();
#pragma unroll
    for (int s = 0; s < 16; ++s) m = fmaxf(m, partial[b * 16 + s]);
    out[(size_t)b * (NC + 1) + NC] = m;
}

extern "C" void kernel_launch(void* const* d_in, const int* in_sizes, int n_in,
                              void* d_out, int out_size, void* d_ws, size_t ws_size,
                              hipStream_t stream) {
    const float* y  = (const float*)d_in[0];   // [512, 1000]
    const float* kW = (const float*)d_in[1];   // [512, 1000]
    float* out = (float*)d_out;                // [512, 1001]

    float* ws      = (float*)d_ws;
    float* kWp     = ws + WS_KWP;
    float* At      = ws + WS_AT;
    float* colnorm = ws + WS_COLNORM;
    float* ymax    = ws + WS_YMAX;
    int*   jidx    = (int*)(ws + WS_JIDX);
    float* partial = ws + WS_PARTIAL;

    lm_prep   <<<dim3(NCP / 256), dim3(256), 0, stream>>>(kW, kWp, colnorm);
    lm_argmax <<<dim3(NB),        dim3(256), 0, stream>>>(y, out, ymax, jidx);
    lm_gatherA<<<dim3(ND * NB / 256), dim3(256), 0, stream>>>(kWp, jidx, At);
    lm_gemm   <<<dim3(NCP / 128, NB / 64), dim3(256), 0, stream>>>(
        At, kWp, colnorm, y, ymax, jidx, partial);
    lm_ybot   <<<dim3(2), dim3(256), 0, stream>>>(partial, out);
}